// WavetableSynth_17016660426917
// MI455X (gfx1250) — compile-verified
//
#include <hip/hip_runtime.h>
#include <hip/hip_bf16.h>

#define SRATE 44100.0f
#define LWT   512
#define NWT   64
#define PH_SCALE (512.0f / 44100.0f)

typedef __attribute__((ext_vector_type(2))) float v2f;
typedef __attribute__((ext_vector_type(8))) float v8f;
typedef int v4i __attribute__((vector_size(16)));

// floored-mod index calc matching jnp: index % L (non-negative), floor/ceil/alpha
__device__ __forceinline__ void idx_calc(float cum, float inc0,
                                         int& il, int& ih, float& alpha) {
    float x = cum - inc0;
    float m = fmodf(x, 512.0f);
    if (m < 0.0f) m += 512.0f;
    float fl = floorf(m);
    alpha = m - fl;
    il = ((int)fl) & 511;
    ih = ((int)ceilf(m)) & 511;
}

// ---------------- k0: transpose wavetables (64,512) -> (512,64) ----------------
__global__ __launch_bounds__(256) void k_transpose(const float* __restrict__ wt,
                                                   float* __restrict__ wtT) {
    int idx = blockIdx.x * 256 + threadIdx.x;          // 0 .. 32767
    int n = idx >> 9, l = idx & 511;
    wtT[l * NWT + n] = wt[idx];
}

// ---------------- k1a: per-256-chunk inclusive scan + chunk totals ----------------
__global__ __launch_bounds__(256) void k_scan_local(const float* __restrict__ pitch,
                                                    float* __restrict__ cum,
                                                    float* __restrict__ part,
                                                    int T, int NCH) {
    __shared__ float s[256];
    int ch = blockIdx.x, b = blockIdx.y, tid = threadIdx.x;
    int t = (ch << 8) + tid;
    float v = (t < T) ? pitch[(size_t)b * T + t] * PH_SCALE : 0.0f;
    s[tid] = v;
    __syncthreads();
#pragma unroll
    for (int off = 1; off < 256; off <<= 1) {
        float add = (tid >= off) ? s[tid - off] : 0.0f;
        __syncthreads();
        s[tid] += add;
        __syncthreads();
    }
    if (t < T) cum[(size_t)b * T + t] = s[tid];
    if (tid == 255) part[b * NCH + ch] = s[255];
}

// ---------------- k1b: exclusive scan of chunk totals (one block per row) ----------------
__global__ __launch_bounds__(256) void k_scan_part(float* __restrict__ part, int NCH) {
    __shared__ float s[256];
    int b = blockIdx.x, tid = threadIdx.x;
    float carry = 0.0f;
    int nloops = (NCH + 255) >> 8;
    for (int ch = 0; ch < nloops; ++ch) {
        int i = (ch << 8) + tid;
        float v = (i < NCH) ? part[b * NCH + i] : 0.0f;
        s[tid] = v;
        __syncthreads();
#pragma unroll
        for (int off = 1; off < 256; off <<= 1) {
            float add = (tid >= off) ? s[tid - off] : 0.0f;
            __syncthreads();
            s[tid] += add;
            __syncthreads();
        }
        if (i < NCH) part[b * NCH + i] = s[tid] - v + carry;   // exclusive
        carry += s[255];
        __syncthreads();
    }
}

// ---------------- k1c: add chunk offsets ----------------
__global__ __launch_bounds__(256) void k_scan_add(float* __restrict__ cum,
                                                  const float* __restrict__ part,
                                                  int T, int NCH) {
    int b = blockIdx.y;
    int t = blockIdx.x * 256 + threadIdx.x;
    if (t < T) cum[(size_t)b * T + t] += part[b * NCH + (t >> 8)];
}

// ---------------- k2: att[t*64+n] = 0.25 * sum_b lerp * y ----------------
// Stages the 128KB transposed table in LDS via CDNA5 async global->LDS loads.
__global__ __launch_bounds__(256) void k_att(const float* __restrict__ cum,
                                             const float* __restrict__ pitch,
                                             const float* __restrict__ y,
                                             const float* __restrict__ wtT,
                                             float* __restrict__ att, int T) {
    __shared__ float wt_s[LWT * NWT];   // 128 KB (<= 320 KB/WGP; 2 blocks resident)

#if defined(__gfx1250__) && __has_builtin(__builtin_amdgcn_global_load_async_to_lds_b128) && \
    __has_builtin(__builtin_amdgcn_s_wait_asynccnt)
    {
        __attribute__((address_space(1))) v4i* g =
            (__attribute__((address_space(1))) v4i*)(wtT);
        __attribute__((address_space(3))) v4i* l =
            (__attribute__((address_space(3))) v4i*)(wt_s);
        // 131072 bytes / (256 threads * 16 B) = 32 iterations, ASYNCcnt-tracked
#pragma unroll 4
        for (int it = 0; it < 32; ++it) {
            int idx = (it << 8) + threadIdx.x;      // v4i element index
            __builtin_amdgcn_global_load_async_to_lds_b128(g + idx, l + idx, 0, 0);
        }
        __builtin_amdgcn_s_wait_asynccnt(0);
    }
#else
    for (int i = threadIdx.x; i < LWT * NWT; i += 256) wt_s[i] = wtT[i];
#endif
    __syncthreads();

    const int n  = threadIdx.x & 63;   // lane group over n -> conflict-free LDS, coalesced att store
    const int tl = threadIdx.x >> 6;   // 4 t per block-iteration
    const int ngroups = (T + 3) >> 2;
    for (int g = blockIdx.x; g < ngroups; g += gridDim.x) {
        int t = (g << 2) + tl;
        if (t >= T) continue;
        float inc0 = pitch[t] * PH_SCALE;
        float acc = 0.0f;
#pragma unroll
        for (int b = 0; b < 4; ++b) {
            int il, ih; float a;
            idx_calc(cum[(size_t)b * T + t], inc0, il, ih, a);
            float yv = y[(size_t)b * T + t];
            float lo = wt_s[il * NWT + n];
            float hi = wt_s[ih * NWT + n];
            acc += (lo + a * (hi - lo)) * yv;
        }
        att[(size_t)t * NWT + n] = 0.25f * acc;
    }
}

// Branch-free boundary load: clamp address into range, select 0 if out of range.
__device__ __forceinline__ float att_at(const float* __restrict__ att,
                                        int col, int n, int T) {
    int colc = min(max(col, 0), T - 1);          // always-valid address
    float v = att[(size_t)colc * NWT + n];       // unconditional coalesced load
    return (col == colc) ? v : 0.0f;             // v_cndmask, no EXEC churn
}

// ---------------- k3: conv (banded GEMM on WMMA) + softmax(n) + weighted output ----------------
// Block = 128 threads (4 waves). Wave w owns n-tile [16w,16w+16), block owns t-tile [16*blk,+16).
// Conv as D(16x16) = A(16x20 att window) x M(20x16 band), via 5x V_WMMA_F32_16X16X4_F32.
__global__ __launch_bounds__(128) void k_conv_softmax_out(
    const float* __restrict__ att,     // (T,64)
    const float* __restrict__ cum,     // (B,T)
    const float* __restrict__ pitch,   // row 0 used
    const float* __restrict__ amp,     // (B,T)
    const float* __restrict__ wtT,     // (512,64)
    const float* __restrict__ conv_w,  // 5
    const float* __restrict__ conv_b,  // 1
    float* __restrict__ out, int T) {
    __shared__ float s_attc[NWT * 16];  // 4 KB: attc then softmax weights, (n, j)

    const int tid  = threadIdx.x;
    const int lane = tid & 31;
    const int n0   = (tid >> 5) << 4;        // wave's n-tile base
    const int tb   = blockIdx.x << 4;        // t-tile base

    const float w0 = conv_w[0], w1 = conv_w[1], w2 = conv_w[2],
                w3 = conv_w[3], w4 = conv_w[4];
    const float cb = conv_b[0];

#if defined(__gfx1250__) && __has_builtin(__builtin_amdgcn_wmma_f32_16x16x4_f32)
    {
        const int m = lane & 15;      // A row (n0+m) / B col j / D col j
        const int h = lane >> 4;      // lane-half: K = r + 2h (ISA 7.12.2)
        const int j = m;
        v8f acc = (v8f)0.0f;
#pragma unroll
        for (int c = 0; c < 5; ++c) { // contraction over 20 shifted columns, 4 at a time
            int col0 = tb - 2 + 4 * c + 2 * h;
            v2f A = {att_at(att, col0,     n0 + m, T),
                     att_at(att, col0 + 1, n0 + m, T)};
            int d0 = 4 * c + 2 * h - j;   // band: M[cc,j] = w[cc-j] if 0<=cc-j<=4
            int d1 = d0 + 1;
            float b0 = (d0 == 0) ? w0 : (d0 == 1) ? w1 : (d0 == 2) ? w2 : (d0 == 3) ? w3 : (d0 == 4) ? w4 : 0.0f;
            float b1 = (d1 == 0) ? w0 : (d1 == 1) ? w1 : (d1 == 2) ? w2 : (d1 == 3) ? w3 : (d1 == 4) ? w4 : 0.0f;
            v2f Bv = {b0, b1};
            acc = __builtin_amdgcn_wmma_f32_16x16x4_f32(
                false, A, false, Bv, (short)0, acc, false, false);
        }
        // D layout: lane<16 -> rows r (VGPR r), lane>=16 -> rows r+8
#pragma unroll
        for (int r = 0; r < 8; ++r)
            s_attc[(n0 + r + 8 * h) * 16 + j] = acc[r] + cb;
    }
#else
    for (int e = tid; e < NWT * 16; e += 128) {
        int n = e >> 4, j = e & 15;
        int t = tb + j;
        float a = 0.0f;
#pragma unroll
        for (int q = 0; q < 5; ++q) {
            float wq = (q == 0) ? w0 : (q == 1) ? w1 : (q == 2) ? w2 : (q == 3) ? w3 : w4;
            a += att_at(att, t + q - 2, n, T) * wq;
        }
        s_attc[n * 16 + j] = a + cb;
    }
#endif
    __syncthreads();

    // softmax over n (64) per column j — tiny, 16 threads handle 16 columns
    if (tid < 16) {
        float mx = -1e30f;
        for (int n = 0; n < NWT; ++n) mx = fmaxf(mx, s_attc[n * 16 + tid]);
        float sum = 0.0f;
        for (int n = 0; n < NWT; ++n) {
            float e = __expf(s_attc[n * 16 + tid] - mx);
            s_attc[n * 16 + tid] = e;
            sum += e;
        }
        float inv = 1.0f / sum;
        for (int n = 0; n < NWT; ++n) s_attc[n * 16 + tid] *= inv;
    }
    __syncthreads();

    // out[b,t] = amp * ((1-a)*dot(soft, wtT[il]) + a*dot(soft, wtT[ih]))
    if (tid < 64) {
        int j = tid & 15, b = tid >> 4;
        int t = tb + j;
        if (t < T) {
            float inc0 = pitch[t] * PH_SCALE;
            int il, ih; float a;
            idx_calc(cum[(size_t)b * T + t], inc0, il, ih, a);
            const float* cl = wtT + il * NWT;
            const float* ch = wtT + ih * NWT;
            float dl = 0.0f, dh = 0.0f;
#pragma unroll 8
            for (int n = 0; n < NWT; ++n) {
                float sw = s_attc[n * 16 + j];
                dl += sw * cl[n];
                dh += sw * ch[n];
            }
            out[(size_t)b * T + t] = amp[(size_t)b * T + t] * ((1.0f - a) * dl + a * dh);
        }
    }
}

extern "C" void kernel_launch(void* const* d_in, const int* in_sizes, int n_in,
                              void* d_out, int out_size, void* d_ws, size_t ws_size,
                              hipStream_t stream) {
    const float* pitch = (const float*)d_in[0];   // (B,T,1)
    const float* amp   = (const float*)d_in[1];   // (B,T,1)
    const float* y     = (const float*)d_in[2];   // (B,T)
    const float* wt    = (const float*)d_in[3];   // (64,512)
    const float* cw    = (const float*)d_in[4];   // (1,1,5)
    const float* cbp   = (const float*)d_in[5];   // (1,)
    const int T   = in_sizes[0] / 4;              // B = 4
    const int NCH = (T + 255) / 256;

    float* ws   = (float*)d_ws;
    float* cum  = ws;                             // 4*T
    float* part = cum + (size_t)4 * T;            // 4*NCH
    float* wtT  = part + (size_t)4 * NCH;         // 512*64
    float* attb = wtT + (size_t)LWT * NWT;        // T*64
    float* out  = (float*)d_out;                  // (B,T,1)

    k_transpose<<<(LWT * NWT) / 256, 256, 0, stream>>>(wt, wtT);

    k_scan_local<<<dim3(NCH, 4), 256, 0, stream>>>(pitch, cum, part, T, NCH);
    k_scan_part<<<4, 256, 0, stream>>>(part, NCH);
    k_scan_add<<<dim3((T + 255) / 256, 4), 256, 0, stream>>>(cum, part, T, NCH);

    int ngroups = (T + 3) / 4;
    int gatt = ngroups < 1024 ? ngroups : 1024;
    k_att<<<gatt, 256, 0, stream>>>(cum, pitch, y, wtT, attb, T);

    k_conv_softmax_out<<<(T + 15) / 16, 128, 0, stream>>>(
        attb, cum, pitch, amp, wtT, cw, cbp, out, T);
}